// STPNR_27023934226877
// MI455X (gfx1250) — compile-verified
//
#include <hip/hip_runtime.h>
#include <math.h>

#define Bdim 32
#define Tdim 64
#define Idim 256
#define Hdim 1024
#define Odim 256
#define Fdim (Hdim + Idim) /* 1280 */

typedef __attribute__((ext_vector_type(2))) float v2f;
typedef __attribute__((ext_vector_type(8))) float v8f;

// ---------------------------------------------------------------------------
// Grid-stride float4 zero fill (b128 stores).
// ---------------------------------------------------------------------------
__global__ void zero_f32_kernel(float4* __restrict__ p, long long n4) {
  long long i = (long long)blockIdx.x * blockDim.x + threadIdx.x;
  const long long stride = (long long)gridDim.x * blockDim.x;
  const float4 z = make_float4(0.f, 0.f, 0.f, 0.f);
  for (; i < n4; i += stride) p[i] = z;
}

// ---------------------------------------------------------------------------
// One recurrent timestep. Block = 256 threads = 8 wave32s; each wave owns one
// (b, h) row of the fast-weight matrix f (1280 fp32 = 40 VGPRs as float4[10]).
// Single pass over f: read once, update once. tin staged in LDS per block.
// Grid: (Hdim/8, Bdim).
// ---------------------------------------------------------------------------
__launch_bounds__(256)
__global__ void stpn_step_kernel(const float* __restrict__ x,     // (B,T,I)
                                 const float* __restrict__ w,     // (H,F)
                                 const float* __restrict__ wl,    // (H,F)
                                 const float* __restrict__ wg,    // (H,F)
                                 const float* __restrict__ bias,  // (H)
                                 const float* __restrict__ h_cur, // (B,H)
                                 float* __restrict__ h_next,      // (B,H)
                                 float* __restrict__ fmat,        // (B,H,F)
                                 int t) {
  __shared__ float tin[Fdim];
  const int b   = blockIdx.y;
  const int tid = threadIdx.x;

  // total_input = concat(x[b,t,:], h[b,:])
  for (int i = tid; i < Fdim; i += 256) {
    tin[i] = (i < Idim) ? x[(size_t)b * (Tdim * Idim) + (size_t)t * Idim + i]
                        : h_cur[(size_t)b * Hdim + (i - Idim)];
  }
  __syncthreads();

  const int wave = tid >> 5;
  const int lane = tid & 31;
  const int hrow = blockIdx.x * 8 + wave;
  const size_t rowoff = (size_t)hrow * Fdim;
  float*       frow = fmat + (size_t)b * Hdim * Fdim + rowoff;
  const float* wrow = w + rowoff;

  // Pass 1: pre = tin . (w + f), ss = |w + f|^2 ; keep f row in registers.
  float4 fkeep[10];
  float pre = 0.f, ss = 0.f;
#pragma unroll
  for (int it = 0; it < 10; ++it) {
    const int idx = it * 128 + lane * 4;
    const float4 f4 = *(const float4*)(frow + idx);
    const float4 w4 = *(const float4*)(wrow + idx);
    const float4 t4 = *(const float4*)(&tin[idx]);
    fkeep[it] = f4;
    const float twx = w4.x + f4.x, twy = w4.y + f4.y;
    const float twz = w4.z + f4.z, tww = w4.w + f4.w;
    pre = fmaf(t4.x, twx, pre); pre = fmaf(t4.y, twy, pre);
    pre = fmaf(t4.z, twz, pre); pre = fmaf(t4.w, tww, pre);
    ss  = fmaf(twx, twx, ss);   ss  = fmaf(twy, twy, ss);
    ss  = fmaf(twz, twz, ss);   ss  = fmaf(tww, tww, ss);
  }
  // wave32 butterfly reduction (all lanes end with the totals)
#pragma unroll
  for (int o = 16; o > 0; o >>= 1) {
    pre += __shfl_xor(pre, o, 32);
    ss  += __shfl_xor(ss,  o, 32);
  }

  const float norm = sqrtf(ss) + 1e-16f;
  const float inv  = 1.0f / norm;
  const float hn   = tanhf(pre * inv + bias[hrow]);
  if (lane == 0) h_next[(size_t)b * Hdim + hrow] = hn;

  // Pass 2: f_new = lambda * (f/norm) + gamma * (tin * h_new)
  const float* lrow = wl + rowoff;
  const float* grow = wg + rowoff;
#pragma unroll
  for (int it = 0; it < 10; ++it) {
    const int idx = it * 128 + lane * 4;
    const float4 l4 = *(const float4*)(lrow + idx);
    const float4 g4 = *(const float4*)(grow + idx);
    const float4 t4 = *(const float4*)(&tin[idx]);
    const float4 f4 = fkeep[it];
    float4 o4;
    o4.x = fmaf(l4.x, f4.x * inv, g4.x * (t4.x * hn));
    o4.y = fmaf(l4.y, f4.y * inv, g4.y * (t4.y * hn));
    o4.z = fmaf(l4.z, f4.z * inv, g4.z * (t4.z * hn));
    o4.w = fmaf(l4.w, f4.w * inv, g4.w * (t4.w * hn));
    *(float4*)(frow + idx) = o4;
  }
}

// ---------------------------------------------------------------------------
// Output projection: tag = hT @ W_out^T + b_out via V_WMMA_F32_16X16X4_F32.
// One wave per 16x16 output tile; grid (B/16, O/16). K loop = 1024 / 4.
// A layout (16x4 f32): lane&15 = M, (lane>>4)*2 = K pair, 2 VGPRs.
// B layout (4x16 f32): lane&15 = N, (lane>>4)*2 = K pair, 2 VGPRs.
// C/D layout: VGPR r -> (M=r, N=lane) lanes 0-15, (M=r+8, N=lane-16) 16-31.
// ---------------------------------------------------------------------------
__launch_bounds__(32)
__global__ void out_proj_wmma_kernel(const float* __restrict__ hT,   // (B,H)
                                     const float* __restrict__ Wout, // (O,H)
                                     const float* __restrict__ bout, // (O)
                                     float* __restrict__ tag) {      // (B,O)
  const int mtile = blockIdx.x;
  const int ntile = blockIdx.y;
  const int lane  = threadIdx.x;
  const int mn    = lane & 15;
  const int khalf = (lane >> 4) << 1;

  const float* arow = hT   + (size_t)(mtile * 16 + mn) * Hdim;
  const float* brow = Wout + (size_t)(ntile * 16 + mn) * Hdim;

  v8f acc = {};
  for (int k = 0; k < Hdim; k += 4) {
    const v2f a  = { arow[k + khalf], arow[k + khalf + 1] };
    const v2f bm = { brow[k + khalf], brow[k + khalf + 1] };
    acc = __builtin_amdgcn_wmma_f32_16x16x4_f32(
        /*neg_a=*/false, a, /*neg_b=*/false, bm,
        /*c_mod=*/(short)0, acc, /*reuse_a=*/false, /*reuse_b=*/false);
  }

  const int n     = ntile * 16 + mn;
  const float bb  = bout[n];
  const int mbase = mtile * 16 + ((lane >> 4) << 3);
#pragma unroll
  for (int r = 0; r < 8; ++r) {
    tag[(size_t)(mbase + r) * Odim + n] = acc[r] + bb;
  }
}

__global__ void copy_f32_kernel(const float* __restrict__ src,
                                float* __restrict__ dst, int n) {
  int i = blockIdx.x * blockDim.x + threadIdx.x;
  if (i < n) dst[i] = src[i];
}

// ---------------------------------------------------------------------------
extern "C" void kernel_launch(void* const* d_in, const int* in_sizes, int n_in,
                              void* d_out, int out_size, void* d_ws, size_t ws_size,
                              hipStream_t stream) {
  const float* x    = (const float*)d_in[0]; // sentence (B,T,I)
  const float* w    = (const float*)d_in[1]; // weight (H,F)
  const float* wl   = (const float*)d_in[2]; // weight_lambda (H,F)
  const float* wg   = (const float*)d_in[3]; // weight_gamma (H,F)
  const float* bias = (const float*)d_in[4]; // (H)
  const float* Wout = (const float*)d_in[5]; // (O,H)
  const float* bout = (const float*)d_in[6]; // (O)

  float* tag = (float*)d_out;           // (B,O)
  float* hT  = tag + Bdim * Odim;       // (B,H)
  float* fT  = hT + Bdim * Hdim;        // (B,H,F) -- used in place as state

  float* hA = (float*)d_ws;             // h ping-pong buffers (2 * B*H fp32)
  float* hB = hA + Bdim * Hdim;

  // f0 = 0 (in place in d_out), h0 = 0 (both ws buffers, contiguous).
  const long long f4n = (long long)Bdim * Hdim * Fdim / 4;
  zero_f32_kernel<<<2048, 256, 0, stream>>>((float4*)fT, f4n);
  zero_f32_kernel<<<64, 256, 0, stream>>>((float4*)hA,
                                          (long long)(2 * Bdim * Hdim) / 4);

  float* hc = hA;
  float* hn = hB;
  const dim3 sgrid(Hdim / 8, Bdim);
  for (int t = 0; t < Tdim; ++t) {
    stpn_step_kernel<<<sgrid, 256, 0, stream>>>(x, w, wl, wg, bias, hc, hn, fT, t);
    float* tmp = hc; hc = hn; hn = tmp;
  }
  // hc now holds h_T.
  out_proj_wmma_kernel<<<dim3(Bdim / 16, Odim / 16), 32, 0, stream>>>(hc, Wout,
                                                                      bout, tag);
  copy_f32_kernel<<<(Bdim * Hdim + 255) / 256, 256, 0, stream>>>(hc, hT,
                                                                 Bdim * Hdim);
}